// SAGE_10900626997366
// MI455X (gfx1250) — compile-verified
//
#include <hip/hip_runtime.h>
#include <stdint.h>

typedef __attribute__((ext_vector_type(2))) float v2f;
typedef __attribute__((ext_vector_type(8))) float v8f;

#define N_NODES 100000
#define D_IN    128
#define LDS_PITCH 132   // 128 + 4 pad: removes 16-way bank conflict, keeps 16B align

// ---------------------------------------------------------------- utilities
__global__ void fill_zero_kernel(float* __restrict__ p, long n) {
  long i = (long)blockIdx.x * blockDim.x + threadIdx.x;
  long stride = (long)gridDim.x * blockDim.x;
  for (; i < n; i += stride) p[i] = 0.0f;
}

__global__ void degree_kernel(const int* __restrict__ dst, int E,
                              float* __restrict__ deg) {
  int e = blockIdx.x * blockDim.x + threadIdx.x;
  if (e < E) atomicAdd(&deg[dst[e]], 1.0f);
}

__global__ void invdeg_kernel(float* __restrict__ deg, int n) {
  int v = blockIdx.x * blockDim.x + threadIdx.x;
  if (v < n) deg[v] = 1.0f / fmaxf(deg[v], 1.0f);
}

// -------------------------------------------------------- neighbor scatter
// One 32-thread group per edge: coalesced 512B row gather (float4 per lane),
// 4 consecutive global_atomic_add_f32 per lane into the destination row.
__global__ void scatter_kernel(const float* __restrict__ h,
                               const int* __restrict__ src,
                               const int* __restrict__ dst, int E,
                               float* __restrict__ agg) {
  long t = (long)blockIdx.x * blockDim.x + threadIdx.x;
  long total = (long)E * 32;
  if (t >= total) return;
  int e = (int)(t >> 5);
  int g = (int)(t & 31);
  const float4* hrow = (const float4*)(h + (long)src[e] * D_IN);
  float4 val = hrow[g];
  float* arow = agg + (long)dst[e] * D_IN + g * 4;
  atomicAdd(arow + 0, val.x);
  atomicAdd(arow + 1, val.y);
  atomicAdd(arow + 2, val.z);
  atomicAdd(arow + 3, val.w);
}

// ------------------------------------------- fused WMMA GEMM, d_out = 128
// One block = one 16-row tile; 8 waves = 8 column tiles of 16.
// A-side tiles (h, agg) staged once per block via global_load_async_to_lds_b128,
// then each wave reads fragments from LDS (ds_load_b64, conflict-free pitch).
__global__ void __launch_bounds__(256)
sage_gemm128_kernel(const float* __restrict__ h,
                    const float* __restrict__ agg,
                    const float* __restrict__ invdeg,
                    const float* __restrict__ Wself,
                    const float* __restrict__ Wneigh,
                    const float* __restrict__ bias,
                    float* __restrict__ out, int do_relu) {
  __shared__ __align__(16) float lds_h[16 * LDS_PITCH];
  __shared__ __align__(16) float lds_a[16 * LDS_PITCH];

  const int row_tile = blockIdx.x;
  const int tid = threadIdx.x;
  const long rowbase = (long)row_tile * 16 * D_IN;

  // ---- async stage: 16 rows x 128 floats of h and agg (16B chunks, 512 each)
#pragma unroll
  for (int p = 0; p < 2; ++p) {
    const int c = tid + p * 256;       // chunk id in [0,512)
    const int r = c >> 5;              // row 0..15
    const int i = c & 31;              // 16B chunk within row
    const float* gph = h + rowbase + r * D_IN + i * 4;
    const float* gpa = agg + rowbase + r * D_IN + i * 4;
    unsigned lofs_h = (unsigned)(uintptr_t)(&lds_h[r * LDS_PITCH + i * 4]);
    unsigned lofs_a = (unsigned)(uintptr_t)(&lds_a[r * LDS_PITCH + i * 4]);
    asm volatile("global_load_async_to_lds_b128 %0, %1, off"
                 :: "v"(lofs_h), "v"(gph) : "memory");
    asm volatile("global_load_async_to_lds_b128 %0, %1, off"
                 :: "v"(lofs_a), "v"(gpa) : "memory");
  }
  asm volatile("s_wait_asynccnt 0x0" ::: "memory");
  __syncthreads();

  // ---- WMMA: lane holds A[M=lane&15][K-pair 2*(lane>>4)], B mirrors it
  const int lane = tid & 31;
  const int wave = tid >> 5;           // column tile 0..7
  const int m = lane & 15;
  const int kb = (lane >> 4) << 1;     // 0 or 2
  const int col = wave * 16 + m;

  const float inv = invdeg[(long)row_tile * 16 + m];
  const float* __restrict__ lh = &lds_h[m * LDS_PITCH];
  const float* __restrict__ la = &lds_a[m * LDS_PITCH];

  v8f acc = {};
  for (int kk = 0; kk < D_IN; kk += 4) {
    const int k0 = kk + kb;
    v2f a_s, a_n, b_s, b_n;
    a_s.x = lh[k0];
    a_s.y = lh[k0 + 1];
    a_n.x = la[k0] * inv;
    a_n.y = la[k0 + 1] * inv;
    b_s.x = Wself[k0 * 128 + col];
    b_s.y = Wself[(k0 + 1) * 128 + col];
    b_n.x = Wneigh[k0 * 128 + col];
    b_n.y = Wneigh[(k0 + 1) * 128 + col];
    acc = __builtin_amdgcn_wmma_f32_16x16x4_f32(false, a_s, false, b_s,
                                                (short)0, acc, false, false);
    acc = __builtin_amdgcn_wmma_f32_16x16x4_f32(false, a_n, false, b_n,
                                                (short)0, acc, false, false);
  }

  // ---- epilogue: D VGPR r holds M = r + 8*(lane>>4), N = lane&15
  const float bval = bias[col];
  const int rbase = (lane >> 4) << 3;  // 0 or 8
  float* optr = out + ((long)row_tile * 16 + rbase) * 128 + col;
#pragma unroll
  for (int r = 0; r < 8; ++r) {
    float v = acc[r] + bval;
    if (do_relu) v = fmaxf(v, 0.0f);
    optr[r * 128] = v;   // immediate-offset stores (r*512B)
  }
}

// ------------------------------------------- fused WMMA GEMM, d_out = 16
// One wave per 16-row tile (single column tile). Register path; W3 is tiny.
__global__ void sage_gemm16_kernel(const float* __restrict__ h,
                                   const float* __restrict__ agg,
                                   const float* __restrict__ invdeg,
                                   const float* __restrict__ Wself,
                                   const float* __restrict__ Wneigh,
                                   const float* __restrict__ bias,
                                   float* __restrict__ out) {
  const int wpb = blockDim.x >> 5;
  const int row_tile = blockIdx.x * wpb + (threadIdx.x >> 5);
  if (row_tile * 16 >= N_NODES) return;  // tile-uniform: EXEC stays full

  const int lane = threadIdx.x & 31;
  const int m = lane & 15;
  const int kb = (lane >> 4) << 1;
  const long row = (long)(row_tile * 16 + m);
  const int col = lane & 15;

  const float inv = invdeg[row];
  const float* __restrict__ hrow = h + row * D_IN;
  const float* __restrict__ arow = agg + row * D_IN;

  v8f acc = {};
  for (int kk = 0; kk < D_IN; kk += 4) {
    const int k0 = kk + kb;
    v2f a_s, a_n, b_s, b_n;
    a_s.x = hrow[k0];
    a_s.y = hrow[k0 + 1];
    a_n.x = arow[k0] * inv;
    a_n.y = arow[k0 + 1] * inv;
    b_s.x = Wself[k0 * 16 + col];
    b_s.y = Wself[(k0 + 1) * 16 + col];
    b_n.x = Wneigh[k0 * 16 + col];
    b_n.y = Wneigh[(k0 + 1) * 16 + col];
    acc = __builtin_amdgcn_wmma_f32_16x16x4_f32(false, a_s, false, b_s,
                                                (short)0, acc, false, false);
    acc = __builtin_amdgcn_wmma_f32_16x16x4_f32(false, a_n, false, b_n,
                                                (short)0, acc, false, false);
  }

  const float bval = bias[col];
  const int rbase = (lane >> 4) << 3;
  float* optr = out + ((long)row_tile * 16 + rbase) * 16 + col;
#pragma unroll
  for (int r = 0; r < 8; ++r) {
    optr[r * 16] = acc[r] + bval;   // no relu on layer 3
  }
}

// --------------------------------------------------------- edge dot scoring
__global__ void edge_score_kernel(const float* __restrict__ h3,
                                  const int* __restrict__ u,
                                  const int* __restrict__ v, int E,
                                  float* __restrict__ out) {
  int e = blockIdx.x * blockDim.x + threadIdx.x;
  if (e >= E) return;
  const float4* a = (const float4*)(h3 + (long)u[e] * 16);
  const float4* b = (const float4*)(h3 + (long)v[e] * 16);
  float s = 0.0f;
#pragma unroll
  for (int i = 0; i < 4; ++i) {
    float4 x = a[i], y = b[i];
    s += x.x * y.x + x.y * y.y + x.z * y.z + x.w * y.w;
  }
  out[e] = s;
}

// -------------------------------------------------------------------- driver
extern "C" void kernel_launch(void* const* d_in, const int* in_sizes, int n_in,
                              void* d_out, int out_size, void* d_ws, size_t ws_size,
                              hipStream_t stream) {
  const float* x       = (const float*)d_in[0];
  const int*   src     = (const int*)d_in[1];
  const int*   dst     = (const int*)d_in[2];
  const int*   pos_u   = (const int*)d_in[3];
  const int*   pos_v   = (const int*)d_in[4];
  const int*   neg_u   = (const int*)d_in[5];
  const int*   neg_v   = (const int*)d_in[6];
  const float* W1_self = (const float*)d_in[7];
  const float* W1_neigh= (const float*)d_in[8];
  const float* b1      = (const float*)d_in[9];
  const float* W2_self = (const float*)d_in[10];
  const float* W2_neigh= (const float*)d_in[11];
  const float* b2      = (const float*)d_in[12];
  const float* W3_self = (const float*)d_in[13];
  const float* W3_neigh= (const float*)d_in[14];
  const float* b3      = (const float*)d_in[15];

  const int E  = in_sizes[1];
  const int Ep = in_sizes[3];
  const int En = in_sizes[5];

  // workspace carve (floats): deg/invdeg | agg | h1 | h2 | h3
  float* deg = (float*)d_ws;
  float* agg = deg + N_NODES;
  float* h1  = agg + (long)N_NODES * D_IN;
  float* h2  = h1  + (long)N_NODES * D_IN;
  float* h3  = h2  + (long)N_NODES * D_IN;

  float* out_pos = (float*)d_out;
  float* out_neg = out_pos + Ep;

  const int edge_blocks   = (E + 255) / 256;
  const long scat_threads = (long)E * 32;
  const int scat_blocks   = (int)((scat_threads + 255) / 256);
  const int row_tiles     = N_NODES / 16;                  // 6250 (exact)
  const int gemm16_blocks = (row_tiles + 7) / 8;           // 8 waves/block

  // degrees -> inverse degrees (shared by all layers)
  fill_zero_kernel<<<1024, 256, 0, stream>>>(deg, (long)N_NODES);
  degree_kernel<<<edge_blocks, 256, 0, stream>>>(dst, E, deg);
  invdeg_kernel<<<(N_NODES + 255) / 256, 256, 0, stream>>>(deg, N_NODES);

  // ---- layer 1: x -> h1 (relu)
  fill_zero_kernel<<<2048, 256, 0, stream>>>(agg, (long)N_NODES * D_IN);
  scatter_kernel<<<scat_blocks, 256, 0, stream>>>(x, src, dst, E, agg);
  sage_gemm128_kernel<<<row_tiles, 256, 0, stream>>>(x, agg, deg, W1_self,
                                                     W1_neigh, b1, h1, 1);

  // ---- layer 2: h1 -> h2 (relu)
  fill_zero_kernel<<<2048, 256, 0, stream>>>(agg, (long)N_NODES * D_IN);
  scatter_kernel<<<scat_blocks, 256, 0, stream>>>(h1, src, dst, E, agg);
  sage_gemm128_kernel<<<row_tiles, 256, 0, stream>>>(h1, agg, deg, W2_self,
                                                     W2_neigh, b2, h2, 1);

  // ---- layer 3: h2 -> h3 (no relu, d_out=16)
  fill_zero_kernel<<<2048, 256, 0, stream>>>(agg, (long)N_NODES * D_IN);
  scatter_kernel<<<scat_blocks, 256, 0, stream>>>(h2, src, dst, E, agg);
  sage_gemm16_kernel<<<gemm16_blocks, 256, 0, stream>>>(h2, agg, deg, W3_self,
                                                        W3_neigh, b3, h3);

  // ---- edge scoring
  edge_score_kernel<<<(Ep + 255) / 256, 256, 0, stream>>>(h3, pos_u, pos_v, Ep, out_pos);
  edge_score_kernel<<<(En + 255) / 256, 256, 0, stream>>>(h3, neg_u, neg_v, En, out_neg);
}